// ResourceEmbedding_83623013253740
// MI455X (gfx1250) — compile-verified
//
#include <hip/hip_runtime.h>
#include <cfloat>

#define R_N   50000
#define O_N   100000
#define EN_N  400000
#define ES_N  400000
#define D_N   128
#define OPF_N 96
#define EA_N  32
#define NB_RED 512

// pair-interleaved transposed-weight LDS layout: 64 K-pairs, stride 258 dwords
#define WI_STRIDE 258
#define WI_FLOATS (64 * WI_STRIDE)  // 16512 floats = 66048 B

typedef float v2f __attribute__((ext_vector_type(2)));
typedef float v8f __attribute__((ext_vector_type(8)));

__device__ __forceinline__ float waveSum(float v) {
#pragma unroll
  for (int off = 16; off > 0; off >>= 1) v += __shfl_xor(v, off, 32);
  return v;
}
__device__ __forceinline__ float leaky(float x) { return x >= 0.0f ? x : 0.2f * x; }
__device__ __forceinline__ float dot4(float4 a, float4 b) {
  return a.x * b.x + a.y * b.y + a.z * b.z + a.w * b.w;
}

// ---------------------------------------------------------------------------
// K0: project attention vectors through the weight matrices (5 matvecs).
// ---------------------------------------------------------------------------
__global__ void k0_prep(const float* __restrict__ Wself, const float* __restrict__ Wres,
                        const float* __restrict__ Wop, const float* __restrict__ aself,
                        const float* __restrict__ aop, const float* __restrict__ ares,
                        float* __restrict__ cvec) {
  const int k = threadIdx.x;  // 0..127
  float cs = 0.f, cu = 0.f, cw = 0.f, cz = 0.f, cv = 0.f;
  for (int j = 0; j < D_N; ++j) {
    const float wsj = Wself[j * D_N + k];
    cs += (aself[j] + aself[D_N + j]) * wsj;
    cu += aop[j] * wsj;
    cw += ares[j] * wsj;
    cz += ares[D_N + j] * Wres[j * D_N + k];
    cv += aop[D_N + j] * Wop[j * D_N + k];
  }
  cvec[k] = cs;
  cvec[128 + k] = cu;
  cvec[256 + k] = cw;
  cvec[384 + k] = cz;
  cvec[512 + k] = cv;
}

// ---------------------------------------------------------------------------
// K1: per-resource scalars. One wave32 per row, float4 per lane.
// ---------------------------------------------------------------------------
__global__ void k1_rows(const float* __restrict__ resources, const float* __restrict__ cvec,
                        float* __restrict__ selfAtt, float* __restrict__ uRes,
                        float* __restrict__ wRes, float* __restrict__ zRes) {
  const int row = (int)((blockIdx.x * blockDim.x + threadIdx.x) >> 5);
  const int lane = threadIdx.x & 31;
  if (row >= R_N) return;
  const float4 r4 = *(const float4*)(resources + row * D_N + lane * 4);
  const float4 cs = *(const float4*)(cvec + lane * 4);
  const float4 cu = *(const float4*)(cvec + 128 + lane * 4);
  const float4 cw = *(const float4*)(cvec + 256 + lane * 4);
  const float4 cz = *(const float4*)(cvec + 384 + lane * 4);
  const float ds = waveSum(dot4(r4, cs));
  const float du = waveSum(dot4(r4, cu));
  const float dw = waveSum(dot4(r4, cw));
  const float dz = waveSum(dot4(r4, cz));
  if (lane == 0) {
    selfAtt[row] = leaky(ds);
    uRes[row] = du;
    wRes[row] = dw;
    zRes[row] = dz;
  }
}

// ---------------------------------------------------------------------------
// K2a: op-edge logits. One wave per edge; lanes 0-23 cover the gathered
// operation row (96), lanes 24-31 cover the edge attrs (32).
// ---------------------------------------------------------------------------
__global__ void k2a_oplogit(const float* __restrict__ ops, const float* __restrict__ attr,
                            const int* __restrict__ nsrc, const int* __restrict__ ndst,
                            const float* __restrict__ cvec, const float* __restrict__ uRes,
                            float* __restrict__ opAtt) {
  const int e = (int)((blockIdx.x * blockDim.x + threadIdx.x) >> 5);
  const int lane = threadIdx.x & 31;
  if (e >= EN_N) return;
  const int s = nsrc[e];
  float4 x, c;
  if (lane < 24) {
    x = *(const float4*)(ops + s * OPF_N + lane * 4);
    c = *(const float4*)(cvec + 512 + lane * 4);
  } else {
    x = *(const float4*)(attr + (size_t)e * EA_N + (lane - 24) * 4);
    c = *(const float4*)(cvec + 512 + OPF_N + (lane - 24) * 4);
  }
  const float p = waveSum(dot4(x, c));
  if (lane == 0) opAtt[e] = leaky(uRes[ndst[e]] + p);
}

// ---------------------------------------------------------------------------
// K2b: res-edge logits. Pure gather, one thread per edge.
// ---------------------------------------------------------------------------
__global__ void k2b_reslogit(const int* __restrict__ ssrc, const int* __restrict__ sdst,
                             const float* __restrict__ wRes, const float* __restrict__ zRes,
                             float* __restrict__ resAtt) {
  const int e = blockIdx.x * blockDim.x + threadIdx.x;
  if (e >= ES_N) return;
  resAtt[e] = leaky(wRes[sdst[e]] + zRes[ssrc[e]]);
}

// ---------------------------------------------------------------------------
// K3: global online-softmax reduction over self_att ++ op_att ++ res_att.
// ---------------------------------------------------------------------------
__global__ void k3a_reduce(const float* __restrict__ selfAtt, const float* __restrict__ opAtt,
                           const float* __restrict__ resAtt, float* __restrict__ redM,
                           float* __restrict__ redS) {
  const int T = R_N + EN_N + ES_N;
  float m = -1e30f, s = 0.f;
  for (int i = blockIdx.x * blockDim.x + threadIdx.x; i < T; i += gridDim.x * blockDim.x) {
    float x;
    if (i < R_N) x = selfAtt[i];
    else if (i < R_N + EN_N) x = opAtt[i - R_N];
    else x = resAtt[i - R_N - EN_N];
    const float M = fmaxf(m, x);
    s = s * __expf(m - M) + __expf(x - M);
    m = M;
  }
  __shared__ float sm[256], ss[256];
  sm[threadIdx.x] = m;
  ss[threadIdx.x] = s;
  __syncthreads();
  for (int off = 128; off > 0; off >>= 1) {
    if (threadIdx.x < off) {
      const float m2 = sm[threadIdx.x + off], s2 = ss[threadIdx.x + off];
      const float M = fmaxf(sm[threadIdx.x], m2);
      ss[threadIdx.x] = ss[threadIdx.x] * __expf(sm[threadIdx.x] - M) + s2 * __expf(m2 - M);
      sm[threadIdx.x] = M;
    }
    __syncthreads();
  }
  if (threadIdx.x == 0) {
    redM[blockIdx.x] = sm[0];
    redS[blockIdx.x] = ss[0];
  }
}

__global__ void k3b_final(const float* __restrict__ redM, const float* __restrict__ redS,
                          float* __restrict__ gstats) {
  const int t = threadIdx.x;  // 256 threads, NB_RED = 512 partials
  float m = redM[t], s = redS[t];
  {
    const float m2 = redM[t + 256], s2 = redS[t + 256];
    const float M = fmaxf(m, m2);
    s = s * __expf(m - M) + s2 * __expf(m2 - M);
    m = M;
  }
  __shared__ float sm[256], ss[256];
  sm[t] = m;
  ss[t] = s;
  __syncthreads();
  for (int off = 128; off > 0; off >>= 1) {
    if (t < off) {
      const float m2 = sm[t + off], s2 = ss[t + off];
      const float M = fmaxf(sm[t], m2);
      ss[t] = ss[t] * __expf(sm[t] - M) + s2 * __expf(m2 - M);
      sm[t] = M;
    }
    __syncthreads();
  }
  if (t == 0) {
    gstats[0] = sm[0];
    gstats[1] = 1.0f / ss[0];
  }
}

// ---------------------------------------------------------------------------
// K4a/K4b: softmax-weighted scatter of RAW input rows (GEMM deferred to K5).
// Accumulators are L2-resident (25.6 MB each); hardware f32 atomics.
// ---------------------------------------------------------------------------
__global__ void k4a_opscatter(const float* __restrict__ ops, const float* __restrict__ attr,
                              const int* __restrict__ nsrc, const int* __restrict__ ndst,
                              const float* __restrict__ opAtt, const float* __restrict__ gstats,
                              float* __restrict__ accOp) {
  const int e = (int)((blockIdx.x * blockDim.x + threadIdx.x) >> 5);
  const int lane = threadIdx.x & 31;
  if (e >= EN_N) return;
  const float n = __expf(opAtt[e] - gstats[0]) * gstats[1];
  const int s = nsrc[e], d = ndst[e];
  float4 x;
  int o;
  if (lane < 24) {
    x = *(const float4*)(ops + s * OPF_N + lane * 4);
    o = lane * 4;
  } else {
    x = *(const float4*)(attr + (size_t)e * EA_N + (lane - 24) * 4);
    o = OPF_N + (lane - 24) * 4;
  }
  float* dst = accOp + d * D_N + o;
  unsafeAtomicAdd(dst + 0, n * x.x);
  unsafeAtomicAdd(dst + 1, n * x.y);
  unsafeAtomicAdd(dst + 2, n * x.z);
  unsafeAtomicAdd(dst + 3, n * x.w);
}

__global__ void k4b_resscatter(const float* __restrict__ resources, const int* __restrict__ ssrc,
                               const int* __restrict__ sdst, const float* __restrict__ resAtt,
                               const float* __restrict__ gstats, float* __restrict__ accRes) {
  const int e = (int)((blockIdx.x * blockDim.x + threadIdx.x) >> 5);
  const int lane = threadIdx.x & 31;
  if (e >= ES_N) return;
  const float n = __expf(resAtt[e] - gstats[0]) * gstats[1];
  const int s = ssrc[e], d = sdst[e];
  const float4 x = *(const float4*)(resources + s * D_N + lane * 4);
  float* dst = accRes + d * D_N + lane * 4;
  unsafeAtomicAdd(dst + 0, n * x.x);
  unsafeAtomicAdd(dst + 1, n * x.y);
  unsafeAtomicAdd(dst + 2, n * x.z);
  unsafeAtomicAdd(dst + 3, n * x.w);
}

// ---------------------------------------------------------------------------
// K5: out = elu(n_self*resources + accOp @ W_op^T + accRes @ W_res^T)
// WMMA f32 16x16x4, K=128, 8 N-tiles per wave, 4 waves (64 rows) per block.
// Two phases (one weight matrix staged at a time): 66 KB LDS -> 4 blocks/WGP.
// Weights stored pair-interleaved so each B fragment is a single ds_load_b64.
// A fragments software-pipelined with unroll capped at 2 to avoid VGPR spills.
// ---------------------------------------------------------------------------
__device__ __forceinline__ void stage_weights_interleaved(const float* __restrict__ W,
                                                          float* __restrict__ Wi, int tid) {
  // thread tid owns column k = tid; reads coalesced, writes conflict-free.
  // Wi[u*WI_STRIDE + n*2 + p] = W[n][2u + p]
  const int u = tid >> 1;
  const int p = tid & 1;
  for (int j = 0; j < D_N; ++j) {
    Wi[u * WI_STRIDE + j * 2 + p] = W[j * D_N + tid];
  }
}

__device__ __forceinline__ void gemm_phase(const float* __restrict__ A, const float* __restrict__ Wi,
                                           int m0, int half, int l15, v8f acc[8]) {
  const float* Arow = A + (size_t)(m0 + l15) * D_N;
  v2f a = *(const v2f*)(Arow + half * 2);
#pragma unroll 2
  for (int kk = 0; kk < 32; ++kk) {
    // prefetch next A fragment (clamped re-load on the last iteration)
    const int kn = (kk < 31) ? kk + 1 : 31;
    const v2f an = *(const v2f*)(Arow + kn * 4 + half * 2);
    const int u = kk * 2 + half;  // K-pair index: covers K = 2u, 2u+1
    const float* base = Wi + u * WI_STRIDE + l15 * 2;
#pragma unroll
    for (int nt = 0; nt < 8; ++nt) {
      const v2f b = *(const v2f*)(base + nt * 32);
      acc[nt] = __builtin_amdgcn_wmma_f32_16x16x4_f32(false, a, false, b, (short)0, acc[nt],
                                                      false, false);
    }
    a = an;
  }
}

__global__ void __launch_bounds__(128, 1)
k5_gemm_elu(const float* __restrict__ accOp, const float* __restrict__ accRes,
            const float* __restrict__ Wop, const float* __restrict__ Wres,
            const float* __restrict__ resources, const float* __restrict__ selfAtt,
            const float* __restrict__ gstats, float* __restrict__ out) {
  extern __shared__ float smem[];  // WI_FLOATS floats (66 KB)
  const int tid = threadIdx.x;     // 128 threads = 4 waves
  const int wave = tid >> 5;
  const int lane = tid & 31;
  const int half = lane >> 4;  // A/B fragment half: 0 -> K=k0,k0+1 ; 1 -> K=k0+2,k0+3
  const int l15 = lane & 15;
  const int m0 = blockIdx.x * 64 + wave * 16;
  const bool active = (m0 < R_N);  // wave-uniform

  v8f acc[8] = {};

  // phase 1: accOp @ Wop^T
  stage_weights_interleaved(Wop, smem, tid);
  __syncthreads();
  if (active) gemm_phase(accOp, smem, m0, half, l15, acc);
  __syncthreads();

  // phase 2: accRes @ Wres^T (accumulate into same tiles)
  stage_weights_interleaved(Wres, smem, tid);
  __syncthreads();
  if (active) gemm_phase(accRes, smem, m0, half, l15, acc);

  if (!active) return;

  // epilogue: + n_self*resources, ELU, store
  const float gmax = gstats[0], invZ = gstats[1];
  float nself[8];
#pragma unroll
  for (int i = 0; i < 8; ++i) {
    const int row = m0 + i + half * 8;
    nself[i] = __expf(selfAtt[row] - gmax) * invZ;
  }
#pragma unroll
  for (int nt = 0; nt < 8; ++nt) {
#pragma unroll
    for (int i = 0; i < 8; ++i) {
      const int row = m0 + i + half * 8;
      const int col = nt * 16 + l15;
      const float x = acc[nt][i] + nself[i] * resources[(size_t)row * D_N + col];
      out[(size_t)row * D_N + col] = x > 0.0f ? x : __expf(x) - 1.0f;
    }
  }
}

// ---------------------------------------------------------------------------
extern "C" void kernel_launch(void* const* d_in, const int* in_sizes, int n_in, void* d_out,
                              int out_size, void* d_ws, size_t ws_size, hipStream_t stream) {
  (void)in_sizes; (void)n_in; (void)out_size; (void)ws_size;
  const float* resources = (const float*)d_in[0];
  const float* operations = (const float*)d_in[1];
  const int* need_src = (const int*)d_in[2];
  const int* need_dst = (const int*)d_in[3];
  const float* need_attr = (const float*)d_in[4];
  const int* same_src = (const int*)d_in[5];
  const int* same_dst = (const int*)d_in[6];
  const float* W_self = (const float*)d_in[7];
  const float* W_res = (const float*)d_in[8];
  const float* W_op = (const float*)d_in[9];
  const float* a_self = (const float*)d_in[10];
  const float* a_op = (const float*)d_in[11];
  const float* a_res = (const float*)d_in[12];
  float* outp = (float*)d_out;

  // workspace layout (floats)
  float* ws = (float*)d_ws;
  float* cvec = ws;                    // 5*128 = 640
  float* gstats = ws + 640;            // 2 (pad to 768)
  float* selfAtt = ws + 768;           // R
  float* uRes = selfAtt + R_N;         // R
  float* wRes = uRes + R_N;            // R
  float* zRes = wRes + R_N;            // R
  float* opAtt = zRes + R_N;           // EN
  float* resAtt = opAtt + EN_N;        // ES
  float* redM = resAtt + ES_N;         // 512
  float* redS = redM + NB_RED;         // 512
  float* accOp = redS + NB_RED;        // offset 1001792 (16B aligned), R*128
  float* accRes = accOp + (size_t)R_N * D_N;

  hipMemsetAsync(accOp, 0, (size_t)2 * R_N * D_N * sizeof(float), stream);

  k0_prep<<<1, 128, 0, stream>>>(W_self, W_res, W_op, a_self, a_op, a_res, cvec);
  k1_rows<<<(R_N + 7) / 8, 256, 0, stream>>>(resources, cvec, selfAtt, uRes, wRes, zRes);
  k2a_oplogit<<<EN_N / 8, 256, 0, stream>>>(operations, need_attr, need_src, need_dst, cvec, uRes,
                                            opAtt);
  k2b_reslogit<<<(ES_N + 255) / 256, 256, 0, stream>>>(same_src, same_dst, wRes, zRes, resAtt);
  k3a_reduce<<<NB_RED, 256, 0, stream>>>(selfAtt, opAtt, resAtt, redM, redS);
  k3b_final<<<1, 256, 0, stream>>>(redM, redS, gstats);
  k4a_opscatter<<<EN_N / 8, 256, 0, stream>>>(operations, need_attr, need_src, need_dst, opAtt,
                                              gstats, accOp);
  k4b_resscatter<<<ES_N / 8, 256, 0, stream>>>(resources, same_src, same_dst, resAtt, gstats,
                                               accRes);
  const size_t ldsBytes = (size_t)WI_FLOATS * sizeof(float);  // 66048 B
  k5_gemm_elu<<<(R_N + 63) / 64, 128, ldsBytes, stream>>>(accOp, accRes, W_op, W_res, resources,
                                                          selfAtt, gstats, outp);
}